// new_GCN_87308095193094
// MI455X (gfx1250) — compile-verified
//
#include <hip/hip_runtime.h>

typedef float v2f __attribute__((ext_vector_type(2)));
typedef float v8f __attribute__((ext_vector_type(8)));

// ---------------------------------------------------------------- utilities
__global__ void fill_f32(float* __restrict__ p, float v, long long n) {
    long long i = (long long)blockIdx.x * blockDim.x + threadIdx.x;
    if (i < n) p[i] = v;
}

__global__ void degree_kernel(const int* __restrict__ col, float* __restrict__ deg, int E) {
    int e = blockIdx.x * blockDim.x + threadIdx.x;
    if (e < E) atomicAdd(&deg[col[e]], 1.0f);
}

__global__ void rsqrt_kernel(const float* __restrict__ deg, float* __restrict__ dinv, int n) {
    int i = blockIdx.x * blockDim.x + threadIdx.x;
    if (i < n) dinv[i] = rsqrtf(deg[i]);
}

// ---------------------------------------------------------------- WMMA GEMM
// C[M x (NT*16)] = A[M x 64] @ B[64 x (NT*16)]  (+ bias[col] if BIAS)
// One wave per 16-row strip; fp32 WMMA 16x16x4, 16 K-steps.
// Fragment layouts per CDNA5 ISA 7.12.2:
//   A 16x4  : lanes 0-15 -> M, vgpr0=K(0|2 by half), vgpr1=K(1|3 by half)
//   B 4x16  : lanes 0-15 -> N, vgpr0=K(0|2 by half), vgpr1=K(1|3 by half)
//   C 16x16 : lanes 0-15 -> N, vgpr v -> M = v (lo half) / v+8 (hi half)
template <int NT, bool BIAS>
__global__ void wmma_gemm_k64(const float* __restrict__ A, const float* __restrict__ B,
                              const float* __restrict__ bias, float* __restrict__ C, int M) {
    const int ldn  = NT * 16;
    const int wave = (int)((blockIdx.x * blockDim.x + threadIdx.x) >> 5);
    const int lane = threadIdx.x & 31;
    const int r0   = wave * 16;
    if (r0 >= M) return;                 // wave-uniform: EXEC stays all-1s for WMMA

    const int half = lane >> 4;          // 0: lanes 0-15, 1: lanes 16-31
    const int lr   = lane & 15;

    v8f acc[NT];
#pragma unroll
    for (int t = 0; t < NT; ++t)
#pragma unroll
        for (int v = 0; v < 8; ++v) acc[t][v] = 0.0f;

    const int arow = min(r0 + lr, M - 1);                    // clamp (M always %16 here)
    const float* __restrict__ ap = A + (size_t)arow * 64 + 2 * half;

    for (int k0 = 0; k0 < 64; k0 += 4) {
        v2f a;
        a.x = ap[k0];
        a.y = ap[k0 + 1];
#pragma unroll
        for (int t = 0; t < NT; ++t) {
            const float* __restrict__ bp = B + (size_t)(k0 + 2 * half) * ldn + t * 16 + lr;
            v2f b;
            b.x = bp[0];
            b.y = bp[ldn];
            acc[t] = __builtin_amdgcn_wmma_f32_16x16x4_f32(
                false, a, false, b, (short)0, acc[t], false, false);
        }
    }

#pragma unroll
    for (int t = 0; t < NT; ++t) {
        const float bv = BIAS ? bias[t * 16 + lr] : 0.0f;
#pragma unroll
        for (int v = 0; v < 8; ++v) {
            const int row = r0 + v + 8 * half;
            if (row < M) C[(size_t)row * ldn + t * 16 + lr] = acc[t][v] + bv;
        }
    }
}

// ---------------------------------------------------------------- edge scatter
// agg[col] += dinv[row]*dinv[col] * h[row]   (16 lanes per edge, float4 each)
__global__ void scatter_edges(const int* __restrict__ row, const int* __restrict__ col,
                              const float* __restrict__ dinv, const float* __restrict__ h,
                              float* __restrict__ agg, int E) {
    long long gid = (long long)blockIdx.x * blockDim.x + threadIdx.x;
    int e = (int)(gid >> 4);
    if (e >= E) return;
    int q = (int)(gid & 15);
    int r = row[e], c = col[e];
    float nrm = dinv[r] * dinv[c];
    float4 v = ((const float4*)(h + (size_t)r * 64))[q];
    float* dst = agg + (size_t)c * 64 + q * 4;
    atomicAdd(dst + 0, nrm * v.x);
    atomicAdd(dst + 1, nrm * v.y);
    atomicAdd(dst + 2, nrm * v.z);
    atomicAdd(dst + 3, nrm * v.w);
}

// ---------------------------------------------------------------- epilogue
// out = agg + g * (1/deg) + b  [; relu]     (float4 over N*16 quads)
template <bool RELU>
__global__ void epilogue_kernel(const float* __restrict__ agg, const float* __restrict__ g,
                                const float* __restrict__ dinv, const float* __restrict__ bias,
                                float* __restrict__ out, long long nquads) {
    long long i = (long long)blockIdx.x * blockDim.x + threadIdx.x;
    if (i >= nquads) return;
    int node = (int)(i >> 4);
    int q    = (int)(i & 15);
    float di = dinv[node];
    float invdeg = di * di;               // dinv = 1/sqrt(deg) -> dinv^2 = 1/deg
    float4 av = ((const float4*)agg)[i];
    float4 gv = ((const float4*)g)[i];
    float4 bv = ((const float4*)bias)[q];
    float4 o;
    o.x = av.x + gv.x * invdeg + bv.x;
    o.y = av.y + gv.y * invdeg + bv.y;
    o.z = av.z + gv.z * invdeg + bv.z;
    o.w = av.w + gv.w * invdeg + bv.w;
    if (RELU) {
        o.x = fmaxf(o.x, 0.0f); o.y = fmaxf(o.y, 0.0f);
        o.z = fmaxf(o.z, 0.0f); o.w = fmaxf(o.w, 0.0f);
    }
    ((float4*)out)[i] = o;
}

// ---------------------------------------------------------------- mean pool
__global__ void pool_kernel(const float* __restrict__ h, const int* __restrict__ batch,
                            float* __restrict__ sums, float* __restrict__ cnts, int N) {
    long long gid = (long long)blockIdx.x * blockDim.x + threadIdx.x;
    int node = (int)(gid >> 4);
    if (node >= N) return;
    int q = (int)(gid & 15);
    int g = batch[node];
    float4 v = ((const float4*)(h + (size_t)node * 64))[q];
    float* dst = sums + (size_t)g * 64 + q * 4;
    atomicAdd(dst + 0, v.x);
    atomicAdd(dst + 1, v.y);
    atomicAdd(dst + 2, v.z);
    atomicAdd(dst + 3, v.w);
    if (q == 0) atomicAdd(&cnts[g], 1.0f);
}

__global__ void pool_div(float* __restrict__ sums, const float* __restrict__ cnts, int n) {
    int i = blockIdx.x * blockDim.x + threadIdx.x;
    if (i < n) sums[i] = sums[i] / fmaxf(cnts[i >> 6], 1.0f);
}

// ---------------------------------------------------------------- launcher
extern "C" void kernel_launch(void* const* d_in, const int* in_sizes, int n_in,
                              void* d_out, int out_size, void* d_ws, size_t ws_size,
                              hipStream_t stream) {
    const float* x    = (const float*)d_in[0];
    const int*   ei   = (const int*)d_in[1];
    const int*   bat  = (const int*)d_in[2];
    const float* W[3] = {(const float*)d_in[3], (const float*)d_in[5], (const float*)d_in[7]};
    const float* b[3] = {(const float*)d_in[4], (const float*)d_in[6], (const float*)d_in[8]};
    const float* Wl   = (const float*)d_in[9];
    const float* bl   = (const float*)d_in[10];

    const int N = in_sizes[0] / 64;        // 100000
    const int E = in_sizes[1] / 2;         // 1600000
    const int G = out_size / 16;           // 512
    const int* rows = ei;
    const int* cols = ei + E;

    float* ws     = (float*)d_ws;
    float* bufG   = ws;                         // N*64  (GEMM output h = in @ W)
    float* bufAgg = bufG   + (size_t)N * 64;    // N*64  (neighbor aggregation)
    float* bufH   = bufAgg + (size_t)N * 64;    // N*64  (layer output)
    float* deg    = bufH   + (size_t)N * 64;    // N
    float* dinv   = deg    + N;                 // N
    float* sums   = dinv   + N;                 // G*64
    float* cnts   = sums   + (size_t)G * 64;    // G
    (void)ws_size; (void)n_in;

    const int T = 256;
    auto blocks = [](long long work, int t) { return (unsigned)((work + t - 1) / t); };

    // degrees (with self loop) and dinv = rsqrt(deg)
    fill_f32<<<blocks(N, T), T, 0, stream>>>(deg, 1.0f, N);
    degree_kernel<<<blocks(E, T), T, 0, stream>>>(cols, deg, E);
    rsqrt_kernel<<<blocks(N, T), T, 0, stream>>>(deg, dinv, N);

    const long long nq = (long long)N * 16;     // float4 quads in an N x 64 buffer
    const float* in = x;
    for (int L = 0; L < 3; ++L) {
        const long long gemm_threads = (long long)(N / 16) * 32;   // one wave / 16 rows
        wmma_gemm_k64<4, false><<<blocks(gemm_threads, T), T, 0, stream>>>(in, W[L], nullptr, bufG, N);
        fill_f32<<<blocks((long long)N * 64, T), T, 0, stream>>>(bufAgg, 0.0f, (long long)N * 64);
        scatter_edges<<<blocks((long long)E * 16, T), T, 0, stream>>>(rows, cols, dinv, bufG, bufAgg, E);
        if (L < 2)
            epilogue_kernel<true><<<blocks(nq, T), T, 0, stream>>>(bufAgg, bufG, dinv, b[L], bufH, nq);
        else
            epilogue_kernel<false><<<blocks(nq, T), T, 0, stream>>>(bufAgg, bufG, dinv, b[L], bufH, nq);
        in = bufH;
    }

    // global mean pool
    fill_f32<<<blocks((long long)G * 64, T), T, 0, stream>>>(sums, 0.0f, (long long)G * 64);
    fill_f32<<<blocks(G, T), T, 0, stream>>>(cnts, 0.0f, G);
    pool_kernel<<<blocks(nq, T), T, 0, stream>>>(bufH, bat, sums, cnts, N);
    pool_div<<<blocks(G * 64, T), T, 0, stream>>>(sums, cnts, G * 64);

    // final linear: [G,64] @ [64,16] + bl, bias fused into WMMA epilogue
    const long long fin_threads = (long long)(G / 16) * 32;
    wmma_gemm_k64<1, true><<<blocks(fin_threads, T), T, 0, stream>>>(sums, Wl, bl, (float*)d_out, G);
}